// OpticalDotProduct_26310969655348
// MI455X (gfx1250) — compile-verified
//
#include <hip/hip_runtime.h>
#include <stdint.h>

#define NROW 8192
#define TPB  256
#define EPT  (NROW / TPB)   // 32 elements per thread
#define RPB  4              // rows per block (pipelined via TDM double-buffer)

// ---- physical constants (folded at compile time, double precision) ----
constexpr double cMZM_LOSS = 0.8912509381337456;   // 10^-0.05
constexpr double cYB_LOSS  = 0.9332543007969911;   // 10^-0.03
constexpr double cMRR_LOSS = 0.8912509381337456;   // 10^-0.05
constexpr double cOPT_GAIN = 10.0;
constexpr double cTIA_GAIN = 100.0;
constexpr double cTHERMAL  = 4.0 * 1.380649e-23 * 300.0 * 1.0e10 / 50.0; // 3.3136e-12
constexpr double cSHOT     = 2.0 * 1.602176634e-19 * 1.0e10;             // 3.2044e-9

typedef unsigned int u32x4 __attribute__((ext_vector_type(4)));
typedef int          i32x8 __attribute__((ext_vector_type(8)));
typedef int          i32x4 __attribute__((ext_vector_type(4)));

#if defined(__has_builtin)
#if __has_builtin(__builtin_amdgcn_tensor_load_to_lds) && __has_builtin(__builtin_amdgcn_s_wait_tensorcnt)
#define HAVE_TDM 1
#endif
#endif

#ifdef HAVE_TDM
// Issue one TDM descriptor: DMA a 32KB row (NROW f32) from global -> LDS.
__device__ __forceinline__ void tdm_load_row(const float* rowptr, float* ldsdst)
{
    unsigned long long ga = (unsigned long long)(uintptr_t)rowptr;
    unsigned int ldsoff = (unsigned int)(size_t)ldsdst;
    u32x4 g0;
    g0[0] = 1u;                                              // count=1, user mode
    g0[1] = ldsoff;                                          // lds_addr (bytes)
    g0[2] = (unsigned int)(ga & 0xFFFFFFFFull);              // global_addr[31:0]
    g0[3] = (unsigned int)((ga >> 32) & 0x1FFFFFFull)        // global_addr[56:32]
            | (2u << 30);                                    // type=2 ("image")
    i32x8 g1;
    g1[0] = (int)(2u << 16);                                 // data_size=4B
    g1[1] = (int)((unsigned)(NROW & 0xFFFF) << 16);          // tensor_dim0 lo16
    g1[2] = (int)(((unsigned)NROW >> 16) | (1u << 16));      // tensor_dim0 hi | tensor_dim1=1
    g1[3] = (int)((unsigned)NROW << 16);                     // tile_dim0 = NROW
    g1[4] = 1;                                               // tile_dim1 = 1
    g1[5] = NROW;                                            // tensor_dim0_stride lo32
    g1[6] = (int)((unsigned)(NROW & 0xFFFF) << 16);          // dim1_stride lo16
    g1[7] = 0;
    i32x4 z4 = {0, 0, 0, 0};
    i32x8 z8 = {0, 0, 0, 0, 0, 0, 0, 0};
    __builtin_amdgcn_tensor_load_to_lds(g0, g1, z4, z4, z8, 0);
}
#endif

// ---------------------------------------------------------------------------
// Main kernel: RPB rows per block; TDM double-buffer pipeline (prefetch row
// k+1 into the other LDS buffer while computing row k), then per-row:
//   pass 1: inorm = max|x|; pass 2: quantize + 5-tap AWG conv + masked reduce.
// ---------------------------------------------------------------------------
__global__ __launch_bounds__(TPB) void optical_main_kernel(
    const float* __restrict__ x,
    const float* __restrict__ scw,
    const float* __restrict__ scale_p,
    const float* __restrict__ ntp, const float* __restrict__ nsp,
    const float* __restrict__ ntn, const float* __restrict__ nsn,
    float* __restrict__ out, int nrows)
{
    __shared__ __align__(16) float ld[2][NROW + 4];  // 2-elem zero halo each side
    __shared__ float redP[8], redN[8], redM[8];
    __shared__ float s_inorm;

    const int t = threadIdx.x;
    const int base = t * EPT;
    const int row0 = blockIdx.x * RPB;

    if (t == 0) {
#pragma unroll
        for (int s = 0; s < 2; ++s) {
            ld[s][0] = 0.0f; ld[s][1] = 0.0f;
            ld[s][NROW + 2] = 0.0f; ld[s][NROW + 3] = 0.0f;
        }
    }

#ifdef HAVE_TDM
    if (t < 32 && row0 < nrows) {                    // wave 0 primes the pipeline
        tdm_load_row(x + (size_t)row0 * NROW, &ld[0][2]);
    }
#endif

    int cur = 0;
    for (int k = 0; k < RPB; ++k) {
        const int b = row0 + k;
        if (b >= nrows) break;                       // block-uniform

#ifdef HAVE_TDM
        if (t < 32) {
            if (k + 1 < RPB && b + 1 < nrows) {
                tdm_load_row(x + (size_t)(b + 1) * NROW, &ld[cur ^ 1][2]);
                __builtin_amdgcn_s_wait_tensorcnt(1);  // row b's DMA complete
            } else {
                __builtin_amdgcn_s_wait_tensorcnt(0);
            }
        }
#else
#pragma unroll
        for (int i = 0; i < EPT; ++i)
            ld[cur][2 + base + i] = x[(size_t)b * NROW + base + i];
#endif
        __syncthreads();

        const float* lrow = &ld[cur][0];

        // ---- pass 1: inorm = max |x| over the row ----
        float m = 0.0f;
        const float4* lp4 = (const float4*)(lrow + base);  // 16B aligned
#pragma unroll
        for (int i = 0; i < 8; ++i) {
            float4 v = lp4[i];
            m = fmaxf(m, fabsf(v.x)); m = fmaxf(m, fabsf(v.y));
            m = fmaxf(m, fabsf(v.z)); m = fmaxf(m, fabsf(v.w));
        }
        if (t == TPB - 1) {  // cover lrow[NROW], lrow[NROW+1]
            m = fmaxf(m, fabsf(lrow[NROW])); m = fmaxf(m, fabsf(lrow[NROW + 1]));
        }
#pragma unroll
        for (int off = 16; off; off >>= 1) m = fmaxf(m, __shfl_xor(m, off, 32));
        if ((t & 31) == 0) redM[t >> 5] = m;
        __syncthreads();
        if (t == 0) {
            float mm = redM[0];
#pragma unroll
            for (int i = 1; i < 8; ++i) mm = fmaxf(mm, redM[i]);
            if (mm <= 1e-9f) mm = 1.0f;
            s_inorm = mm;
        }
        __syncthreads();
        const float inorm = s_inorm;
        const float qs = 255.0f / inorm;

        // ---- pass 2: quantize + sliding-window conv + masked reduce ----
        auto qd = [&](float xv) -> float {
            float d = __builtin_rintf(fabsf(xv) * qs) * (1.0f / 255.0f);
            return (xv > 0.0f) ? d : -d;   // sign bit carries t_pos
        };
        float w0 = qd(lrow[base + 0]);
        float w1 = qd(lrow[base + 1]);
        float w2 = qd(lrow[base + 2]);
        float w3 = qd(lrow[base + 3]);
        float pos = 0.0f, neg = 0.0f;
        const float4* sw4 = (const float4*)(scw + base);
#pragma unroll
        for (int g = 0; g < 8; ++g) {
            float4 sv = sw4[g];
            float swa[4] = {sv.x, sv.y, sv.z, sv.w};
#pragma unroll
            for (int c = 0; c < 4; ++c) {
                int j = base + g * 4 + c;
                float w4v = qd(lrow[j + 4]);
                float l = fmaf(fabsf(w1) + fabsf(w3), 0.01f,
                          fmaf(fabsf(w0) + fabsf(w4v), 0.0025f, fabsf(w2)));
                float s = swa[c];
                float v = l * fabsf(s);
                unsigned df = (__float_as_uint(w2) ^ __float_as_uint(s)) & 0x80000000u;
                pos += df ? 0.0f : v;
                neg += df ? v : 0.0f;
                w0 = w1; w1 = w2; w2 = w3; w3 = w4v;
            }
        }
#pragma unroll
        for (int off = 16; off; off >>= 1) {
            pos += __shfl_xor(pos, off, 32);
            neg += __shfl_xor(neg, off, 32);
        }
        if ((t & 31) == 0) { redP[t >> 5] = pos; redN[t >> 5] = neg; }
        __syncthreads();

        if (t == 0) {
            float P = 0.0f, Nn = 0.0f;
#pragma unroll
            for (int i = 0; i < 8; ++i) { P += redP[i]; Nn += redN[i]; }
            const float thermal = (float)cTHERMAL, shot = (float)cSHOT;
            float tp = P + 1e-12f;            // photodiode +dark
            tp += ntp[b] * thermal;
            tp *= (1.0f + nsp[b] * shot);
            float tn = Nn + 1e-12f;
            tn += ntn[b] * thermal;
            tn *= (1.0f + nsn[b] * shot);
            float cur_i = tp - tn;
            float v   = fabsf(cur_i * 100.0f);                     // TIA
            float sgn = (cur_i >= 0.0f) ? 1.0f : -1.0f;
            float va  = __builtin_rintf(fminf(v, 1.0f) * 255.0f) * (1.0f / 255.0f);
            out[b] = va * sgn * scale_p[0] * inorm;
        }
        __syncthreads();    // all reads of ld[cur] done before it is reused
        cur ^= 1;
    }
}

// ---------------------------------------------------------------------------
// Prep kernel: wnorm = max|w|; scw[j] = sign(w[j]) * OPT*YB*MZM*MRR*w_eff(j)*|w[j]|/wnorm
// ws layout: [0..NROW-1] = scw, [NROW] = final output scale
// ---------------------------------------------------------------------------
__global__ __launch_bounds__(TPB) void optical_prep_kernel(
    const float* __restrict__ w, float* __restrict__ ws)
{
    __shared__ float red[8];
    __shared__ float s_wnorm;
    const int t = threadIdx.x;
    const int base = t * EPT;

    float4 buf[8];
    const float4* w4 = (const float4*)(w + base);
#pragma unroll
    for (int i = 0; i < 8; ++i) buf[i] = w4[i];

    float m = 0.0f;
#pragma unroll
    for (int i = 0; i < 8; ++i) {
        m = fmaxf(m, fabsf(buf[i].x)); m = fmaxf(m, fabsf(buf[i].y));
        m = fmaxf(m, fabsf(buf[i].z)); m = fmaxf(m, fabsf(buf[i].w));
    }
#pragma unroll
    for (int off = 16; off; off >>= 1) m = fmaxf(m, __shfl_xor(m, off, 32));
    if ((t & 31) == 0) red[t >> 5] = m;
    __syncthreads();
    if (t == 0) {
        float mm = red[0];
#pragma unroll
        for (int i = 1; i < 8; ++i) mm = fmaxf(mm, red[i]);
        if (mm <= 1e-9f) mm = 1.0f;
        s_wnorm = mm;
    }
    __syncthreads();
    const float wnorm = s_wnorm;
    const float inv_wnorm = 1.0f / wnorm;
    const float cbase = (float)(cOPT_GAIN * cYB_LOSS * cMZM_LOSS * cMRR_LOSS);

#pragma unroll
    for (int i = 0; i < 8; ++i) {
        float comp[4] = {buf[i].x, buf[i].y, buf[i].z, buf[i].w};
#pragma unroll
        for (int c = 0; c < 4; ++c) {
            int j = base + i * 4 + c;
            float we = 1.025f;                       // interior MRR kernel sum
            if (j == 0 || j == NROW - 1) we = 1.0125f;
            else if (j == 1 || j == NROW - 2) we = 1.0225f;
            float wj = comp[c];
            float cw = fabsf(wj) * inv_wnorm * cbase * we;
            ws[j] = (wj >= 0.0f) ? cw : -cw;         // sign bit encodes w_pos
        }
    }
    if (t == 0) {
        ws[NROW] = wnorm / (float)(cOPT_GAIN * cTIA_GAIN * cMRR_LOSS * cYB_LOSS * cMZM_LOSS);
    }
}

extern "C" void kernel_launch(void* const* d_in, const int* in_sizes, int n_in,
                              void* d_out, int out_size, void* d_ws, size_t ws_size,
                              hipStream_t stream) {
    const float* x   = (const float*)d_in[0];
    const float* w   = (const float*)d_in[1];
    const float* ntp = (const float*)d_in[2];
    const float* nsp = (const float*)d_in[3];
    const float* ntn = (const float*)d_in[4];
    const float* nsn = (const float*)d_in[5];
    float* out = (float*)d_out;
    float* ws  = (float*)d_ws;

    optical_prep_kernel<<<1, TPB, 0, stream>>>(w, ws);
    const int nblk = (out_size + RPB - 1) / RPB;
    optical_main_kernel<<<nblk, TPB, 0, stream>>>(
        x, ws, ws + NROW, ntp, nsp, ntn, nsn, out, out_size);
}